// GGNPoolingLayer_67276367724845
// MI455X (gfx1250) — compile-verified
//
#include <hip/hip_runtime.h>

// ---------------------------------------------------------------------------
// GGN pooling layer for MI455X (gfx1250, wave32).
// Distance GEMM (K=3, padded to 4) runs on the matrix pipe via
// V_WMMA_F32_16X16X4_F32; masked min-reduce runs on VALU and co-executes
// with WMMA; hash matching is a ballot-based any-reduce whose accumulation
// is scalar (SALU) work. Candidates are staged in LDS pre-swizzled into the
// B-fragment layout so each tile needs a single unconditional ds_load_b64.
// ---------------------------------------------------------------------------

#define HASH_BASE 100000

namespace {
constexpr int kB  = 2;
constexpr int kV  = 8;
constexpr int kG  = 64 * 64;       // 4096
constexpr int kC  = 64;
constexpr int kVG = kV * kG;       // 32768
constexpr float kLamSq = 0.01f;    // 0.1^2

constexpr int QPW = 16;            // queries per wave (one WMMA M-tile)
constexpr int WPB = 8;             // waves per block (256 threads)
constexpr int QPB = QPW * WPB;     // 128 queries per block
constexpr int CCH = 256;           // candidates staged in LDS per chunk
}

typedef __attribute__((ext_vector_type(2))) float v2f;
typedef __attribute__((ext_vector_type(8))) float v8f;

// ---------------------------------------------------------------------------
__global__ void ggn_init_keep(int* __restrict__ keep) {
  int i = blockIdx.x * 256 + threadIdx.x;
  if (i < kB * kVG) keep[i] = 1;
}

// ---------------------------------------------------------------------------
// Connected-component labels via min-propagation (tiny: B*V = 16 work items).
__global__ void ggn_labels(const int* __restrict__ A, int* __restrict__ labels) {
  __shared__ int sl[kB * kV];
  int tid = threadIdx.x;
  int b = tid / kV, v = tid % kV;
  bool act = tid < kB * kV;
  if (act) sl[tid] = v;
  __syncthreads();
  for (int it = 0; it < kV; ++it) {
    int nl = kV;
    if (act) {
      for (int u = 0; u < kV; ++u) {
        bool adj = (u == v) || (A[(b * kV + v) * kV + u] > 0) ||
                   (A[(b * kV + u) * kV + v] > 0);
        if (adj) nl = min(nl, sl[b * kV + u]);
      }
    }
    __syncthreads();
    if (act) sl[tid] = min(sl[tid], nl);
    __syncthreads();
  }
  if (act) labels[tid] = sl[tid];
}

// ---------------------------------------------------------------------------
// Phase kernel for view v: for each query point g of view v, find
//   min over candidates of (|q|^2 + |r|^2 - 2 q.r)  (masked by keep & prior)
//   and any hash-id match, then update keep[b][v][g].
// One wave owns 16 queries; candidates streamed through LDS in chunks of 256.
__global__ __launch_bounds__(256)
void ggn_phase(const float* __restrict__ means,   // (B,V,G,3)
               const int*   __restrict__ xy,      // (B,V,V,G,2)
               const int*   __restrict__ labels,  // (B,V)
               int*         __restrict__ keep,    // (B,V,G)
               int v) {
  __shared__ __align__(16) float sQ4[QPB][4];   // {x,y,z,0} A-fragment layout
  __shared__ float sQn[QPB];
  __shared__ int   sCid[QPB];
  __shared__ __align__(16) float sPos[CCH * 3]; // flat coalesced staging
  __shared__ __align__(16) float sC4[CCH][4];   // {x,y,z,0} B-fragment layout
  __shared__ float sRn[CCH];   // keep ? |r|^2 : +inf   (mask folded in)
  __shared__ int   sId[CCH];   // keep ? hash id : -1   (-1 never matches)
  __shared__ int   sLab[kV];

  const int b    = blockIdx.y;
  const int qblk = blockIdx.x * QPB;
  const int tid  = threadIdx.x;
  const int lane = tid & 31;
  const int wid  = tid >> 5;
  const int m    = lane & 15;
  const bool hi  = lane >= 16;

  if (tid < kV) sLab[tid] = labels[b * kV + tid];
  __syncthreads();
  const int labv = sLab[v];
  if (labv == v) return;  // is_first: keep stays 1 (set by init kernel)

  // Stage this block's 128 query points (padded), norms and current hash ids.
  for (int i = tid; i < QPB; i += 256) {
    int g = qblk + i;
    const float* mp = means + ((size_t)(b * kV + v) * kG + g) * 3;
    float x = mp[0], y = mp[1], z = mp[2];
    sQ4[i][0] = x; sQ4[i][1] = y; sQ4[i][2] = z; sQ4[i][3] = 0.0f;
    sQn[i] = x * x + y * y + z * z;
    const int2 ipv =
        *(const int2*)(xy + ((((size_t)b * kV + v) * kV + v) * kG + g) * 2);
    sCid[i] = ipv.x * HASH_BASE + ipv.y;
  }
  __syncthreads();

  // A fragment: 16x4 f32 (K padded 3->4 with zero); one ds_load_b64 per lane.
  const int qoff = wid * QPW;
  const float2 av = *(const float2*)&sQ4[qoff + m][hi ? 2 : 0];
  v2f afrag; afrag.x = av.x; afrag.y = av.y;

  // Query ids for this wave, kept in registers (ds_load_b128 x4 at stage).
  int qid[16];
#pragma unroll
  for (int mm = 0; mm < 16; ++mm) qid[mm] = sCid[qoff + mm];

  float minacc[8];
#pragma unroll
  for (int r = 0; r < 8; ++r) minacc[r] = __builtin_inff();
  unsigned matched = 0u;  // wave-uniform (accumulated from ballots)

  for (int u = 0; u < v; ++u) {
    if (sLab[u] != labv) continue;  // uniform branch
    const float* mu = means + (size_t)(b * kV + u) * kG * 3;
    const int*   ku = keep  + (size_t)(b * kV + u) * kG;
    const int*   xu = xy + (((size_t)b * kV + u) * kV + v) * kG * 2;

    for (int base = 0; base < kG; base += CCH) {
      __syncthreads();
      // Coalesced 128-bit staging of candidate positions (768 floats).
      if (tid < (CCH * 3) / 4) {
        float4 p = *(const float4*)(mu + (size_t)base * 3 + tid * 4);
        *(float4*)&sPos[tid * 4] = p;
      }
      // Prefetch next chunk's positions while we crunch this one.
      if (base + CCH < kG && tid < (CCH * 3) / 4)
        __builtin_prefetch(mu + (size_t)(base + CCH) * 3 + tid * 4, 0, 0);
      __syncthreads();
      // Repack into padded B-fragment layout with mask folded into rn/id.
      for (int j = tid; j < CCH; j += 256) {
        float x = sPos[j * 3 + 0], y = sPos[j * 3 + 1], z = sPos[j * 3 + 2];
        int kp = ku[base + j];
        int2 xv = *(const int2*)(xu + (size_t)(base + j) * 2);
        sC4[j][0] = x; sC4[j][1] = y; sC4[j][2] = z; sC4[j][3] = 0.0f;
        sRn[j] = kp ? (x * x + y * y + z * z) : __builtin_inff();
        sId[j] = kp ? (xv.x * HASH_BASE + xv.y) : -1;
      }
      __syncthreads();

      // Distance tiles: one V_WMMA_F32_16X16X4_F32 per 16 candidates.
      for (int t = 0; t < CCH; t += 16) {
        int n = t + m;
        const float2 bv = *(const float2*)&sC4[n][hi ? 2 : 0];
        v2f bfrag; bfrag.x = bv.x; bfrag.y = bv.y;
        v8f c = {};
        c = __builtin_amdgcn_wmma_f32_16x16x4_f32(
            /*neg_a=*/false, afrag, /*neg_b=*/false, bfrag,
            /*c_mod=*/(short)0, c, /*reuse_a=*/false, /*reuse_b=*/false);
        float rnm = sRn[n];  // +inf for masked candidates
#pragma unroll
        for (int r = 0; r < 8; ++r) {
          float val = fmaf(-2.0f, c[r], rnm);  // |r|^2 - 2 q.r
          minacc[r] = fminf(minacc[r], val);
        }
      }

      // Hash-id match: ballot any-reduce. One v_cmp covers 32 candidates per
      // query; the accumulate is scalar (SALU) and matched is wave-uniform.
      for (int j0 = 0; j0 < CCH; j0 += 32) {
        int idm = sId[j0 + lane];  // -1 for masked: never matches
#pragma unroll
        for (int mm = 0; mm < 16; ++mm)
          matched |= (__ballot(idm == qid[mm]) != 0ull) ? (1u << mm) : 0u;
      }
    }
  }

  // Min across each 16-lane half (N dimension); matched is already uniform.
#pragma unroll
  for (int r = 0; r < 8; ++r) {
#pragma unroll
    for (int off = 8; off >= 1; off >>= 1)
      minacc[r] = fminf(minacc[r], __shfl_xor(minacc[r], off, 32));
  }

  if (m == 0) {  // lanes 0 (rows 0..7) and 16 (rows 8..15)
    int rowbase = hi ? 8 : 0;
    for (int r = 0; r < 8; ++r) {
      int row = rowbase + r;
      float md = minacc[r] + sQn[qoff + row];
      bool dist_ok = md <= kLamSq;
      bool mt = (matched >> row) & 1u;
      keep[((size_t)(b * kV + v)) * kG + qblk + qoff + row] =
          (dist_ok && mt) ? 0 : 1;
    }
  }
}

// ---------------------------------------------------------------------------
// One wave per batch: stable exclusive prefix sum over keep flags (ballot).
__global__ void ggn_scan(const int* __restrict__ keep, int* __restrict__ rank,
                         int* __restrict__ count) {
  int b = blockIdx.x;
  int lane = threadIdx.x;
  const int* kf = keep + (size_t)b * kVG;
  int* rf = rank + (size_t)b * kVG;
  unsigned ltmask = (1u << lane) - 1u;
  int running = 0;
  for (int base = 0; base < kVG; base += 32) {
    int f = kf[base + lane] ? 1 : 0;
    unsigned m32 = (unsigned)__ballot(f);
    rf[base + lane] = running + __popc(m32 & ltmask);
    running += __popc(m32);
  }
  if (lane == 0) count[b] = running;
}

// ---------------------------------------------------------------------------
// Zero pf/pm regions and emit the keep mask (as float) in one pass.
__global__ void ggn_fill(const int* __restrict__ keep, float* __restrict__ out) {
  size_t i = (size_t)blockIdx.x * 256 + threadIdx.x;
  const size_t pfpm = (size_t)kB * kVG * (kC + 3);
  const size_t total = pfpm + (size_t)kB * kVG;
  if (i >= total) return;
  if (i < pfpm) out[i] = 0.0f;
  else          out[i] = keep[i - pfpm] ? 1.0f : 0.0f;
}

// ---------------------------------------------------------------------------
// Scatter kept rows into compacted positions (64 threads per source row).
__global__ __launch_bounds__(256)
void ggn_scatter(const float* __restrict__ features,  // (B,VG,C)
                 const float* __restrict__ means,     // (B,VG,3)
                 const int*   __restrict__ keep,
                 const int*   __restrict__ rank,
                 float*       __restrict__ out) {
  int b = blockIdx.y;
  int tid = threadIdx.x;
  size_t i = (size_t)blockIdx.x * 4 + (tid >> 6);
  int c = tid & 63;
  if (!keep[(size_t)b * kVG + i]) return;
  int r = rank[(size_t)b * kVG + i];
  float* pf = out;
  float* pm = out + (size_t)kB * kVG * kC;
  pf[((size_t)b * kVG + r) * kC + c] = features[((size_t)b * kVG + i) * kC + c];
  if (c < 3)
    pm[((size_t)b * kVG + r) * 3 + c] = means[((size_t)b * kVG + i) * 3 + c];
}

// ---------------------------------------------------------------------------
extern "C" void kernel_launch(void* const* d_in, const int* in_sizes, int n_in,
                              void* d_out, int out_size, void* d_ws,
                              size_t ws_size, hipStream_t stream) {
  const float* features = (const float*)d_in[0];
  const float* means    = (const float*)d_in[1];
  const int*   xy       = (const int*)d_in[2];
  const int*   A        = (const int*)d_in[3];
  float* out = (float*)d_out;

  char* ws = (char*)d_ws;
  int* labels = (int*)ws;                                       // B*V
  int* keep   = (int*)(ws + 256);                               // B*V*G
  int* rank   = (int*)(ws + 256 + sizeof(int) * kB * kVG);      // B*VG
  int* count  = (int*)(ws + 256 + 2 * sizeof(int) * (size_t)kB * kVG);

  ggn_init_keep<<<(kB * kVG + 255) / 256, 256, 0, stream>>>(keep);
  ggn_labels<<<1, 32, 0, stream>>>(A, labels);

  for (int v = 1; v < kV; ++v)
    ggn_phase<<<dim3(kG / QPB, kB), 256, 0, stream>>>(means, xy, labels, keep, v);

  ggn_scan<<<kB, 32, 0, stream>>>(keep, rank, count);

  const size_t total_fill = (size_t)kB * kVG * (kC + 3 + 1);
  ggn_fill<<<(unsigned)((total_fill + 255) / 256), 256, 0, stream>>>(keep, out);

  ggn_scatter<<<dim3(kVG / 4, kB), 256, 0, stream>>>(features, means, keep,
                                                     rank, out);
}